// LegacySpikeGeneratorBinomial_63385127354876
// MI455X (gfx1250) — compile-verified
//
#include <hip/hip_runtime.h>
#include <hip/hip_bf16.h>
#include <stdint.h>

// ---------------------------------------------------------------------------
// LegacySpikeGeneratorBinomial: counts[i] = sum_{d=0..9} 1[p_i + s*z_{i,d} > u_{i,d}]
//   p_i = rate_i / 10000, s = 0.001, z ~ N(0,1) (Box-Muller), u ~ U(0,1).
// RNG/transcendental bound. PCG-hash uniforms (3 hashes per draw pair: 24-bit
// Box-Muller inputs + one hash split into two 16-bit thresholds), raw hardware
// v_sin/v_cos (revolution-native: sin(2*pi*b) == v_sin(b)), v_log_f32 (log2)
// with ln2 folded, raw v_sqrt_f32.
// gfx1250 path: double-buffered GLOBAL_LOAD_ASYNC_TO_LDS_B128 prefetch with
// s_wait_asynccnt, b128 vector loads/stores.
// ---------------------------------------------------------------------------

#define BLOCK 128          // 4 waves (wave32)
#define MAXGRID 2048
#define SR_INV 1.0e-4f
#define SIGMA  1.0e-3f
#define TWO_PI 6.283185307179586f
#define NEG_2LN2 -1.3862943611198906f   // -2*ln(2): -2*ln(a) = NEG_2LN2*log2(a)

#if defined(__has_builtin)
#if __has_builtin(__builtin_amdgcn_global_load_async_to_lds_b128) && \
    __has_builtin(__builtin_amdgcn_s_wait_asynccnt)
#define USE_ASYNC_LDS 1
#endif
#if __has_builtin(__builtin_amdgcn_sinf) && __has_builtin(__builtin_amdgcn_cosf)
#define USE_RAW_SINCOS 1
#endif
#if __has_builtin(__builtin_amdgcn_logf)
#define USE_RAW_LOG 1
#endif
#if __has_builtin(__builtin_amdgcn_sqrtf)
#define USE_RAW_SQRT 1
#endif
#endif
#ifndef USE_ASYNC_LDS
#define USE_ASYNC_LDS 0
#endif

__device__ __forceinline__ unsigned pcg_hash(unsigned x) {
  x = x * 747796405u + 2891336453u;
  unsigned w = ((x >> ((x >> 28u) + 4u)) ^ x) * 277803737u;
  return (w >> 22u) ^ w;
}

__device__ __forceinline__ float u01_24(unsigned h) {  // 24-bit uniform [0,1)
  return (float)(h >> 8) * 0x1.0p-24f;
}

// sin(2*pi*x), cos(2*pi*x) for x in [0,1): v_sin/v_cos are revolution-native.
__device__ __forceinline__ float sin2pi(float x) {
#if defined(USE_RAW_SINCOS)
  return __builtin_amdgcn_sinf(x);
#else
  return __sinf(TWO_PI * x);
#endif
}
__device__ __forceinline__ float cos2pi(float x) {
#if defined(USE_RAW_SINCOS)
  return __builtin_amdgcn_cosf(x);
#else
  return __cosf(TWO_PI * x);
#endif
}
__device__ __forceinline__ float log2_raw(float x) {
#if defined(USE_RAW_LOG)
  return __builtin_amdgcn_logf(x);   // v_log_f32 = log2(x)
#else
  return __log2f(x);
#endif
}
__device__ __forceinline__ float sqrt_raw(float x) {
#if defined(USE_RAW_SQRT)
  return __builtin_amdgcn_sqrtf(x);  // v_sqrt_f32
#else
  return __fsqrt_rn(x);
#endif
}

// 10 Bernoulli draws with Gaussian-perturbed threshold; 5 Box-Muller pairs,
// 3 PCG hashes per pair (15 per element).
__device__ __forceinline__ float draw10(float p, unsigned base) {
  float cnt = 0.0f;
#pragma unroll
  for (int pr = 0; pr < 5; ++pr) {
    const unsigned s = base + 3u * (unsigned)pr;
    const unsigned hu = pcg_hash(s + 2u);
    const float a = u01_24(pcg_hash(s + 0u)) + 0x1.0p-25f;  // (0,1], log-safe
    const float b = u01_24(pcg_hash(s + 1u));
    const float r = sqrt_raw(NEG_2LN2 * log2_raw(a));       // sqrt(-2 ln a)
    const float u0 = (float)(hu & 0xffffu) * 0x1.0p-16f;    // 16-bit thresholds
    const float u1 = (float)(hu >> 16u) * 0x1.0p-16f;
    cnt += (fmaf(SIGMA * r, cos2pi(b), p) > u0) ? 1.0f : 0.0f;
    cnt += (fmaf(SIGMA * r, sin2pi(b), p) > u1) ? 1.0f : 0.0f;
  }
  return cnt;
}

#if USE_ASYNC_LDS
typedef int v4i_t __attribute__((vector_size(4 * sizeof(int))));
typedef __attribute__((address_space(1))) v4i_t* g_v4i_p;  // global (__device__)
typedef __attribute__((address_space(3))) v4i_t* l_v4i_p;  // LDS (__shared__)

__device__ __forceinline__ void async_cp16(const float4* g, float4* l) {
  // Integer-detour address-space casts (generic flat -> AS1 / AS3).
  // Low 32 bits of a flat LDS address are the LDS byte offset.
  g_v4i_p gp = (g_v4i_p)(unsigned long long)(uintptr_t)g;
  l_v4i_p lp = (l_v4i_p)(unsigned)(uintptr_t)l;
  __builtin_amdgcn_global_load_async_to_lds_b128(gp, lp, /*offset=*/0, /*cpol=*/0);
}
#endif

__global__ __launch_bounds__(BLOCK) void spike_binomial_kernel(
    const float* __restrict__ in, float* __restrict__ out, int n, int n4) {
  const float4* __restrict__ gin = reinterpret_cast<const float4*>(in);
  float4* __restrict__ gout = reinterpret_cast<float4*>(out);
  const int tid = threadIdx.x;
  const int ntiles = n4 / BLOCK;

#if USE_ASYNC_LDS
  __shared__ float4 stage[2][BLOCK];  // 4 KB double buffer, per-thread slots
#endif

  int tile = (int)blockIdx.x;
  int bi = 0;

#if USE_ASYNC_LDS
  if (tile < ntiles) {
    async_cp16(&gin[(size_t)tile * BLOCK + tid], &stage[0][tid]);
  }
#endif

  while (tile < ntiles) {
    const int next = tile + (int)gridDim.x;
    float4 v;
#if USE_ASYNC_LDS
    if (next < ntiles) {
      async_cp16(&gin[(size_t)next * BLOCK + tid], &stage[bi ^ 1][tid]);
      __builtin_amdgcn_s_wait_asynccnt(1);  // current tile's copy is done
    } else {
      __builtin_amdgcn_s_wait_asynccnt(0);
    }
    v = stage[bi][tid];  // ds_load_b128; dscnt wait protects buffer reuse (WAR)
#else
    v = gin[(size_t)tile * BLOCK + tid];
#endif
    const unsigned e = ((unsigned)tile * BLOCK + (unsigned)tid) * 4u;
    float4 o;
    o.x = draw10(v.x * SR_INV, (e + 0u) * 16u);
    o.y = draw10(v.y * SR_INV, (e + 1u) * 16u);
    o.z = draw10(v.z * SR_INV, (e + 2u) * 16u);
    o.w = draw10(v.w * SR_INV, (e + 3u) * 16u);
    gout[(size_t)tile * BLOCK + tid] = o;  // global_store_b128
    bi ^= 1;
    tile = next;
  }

  // Ragged epilogue (n = 8,000,000 divides exactly here, but stay generic).
  if (blockIdx.x == 0) {
    for (int i = ntiles * BLOCK + tid; i < n4; i += BLOCK) {
      float4 v = gin[i];
      const unsigned e = (unsigned)i * 4u;
      float4 o;
      o.x = draw10(v.x * SR_INV, (e + 0u) * 16u);
      o.y = draw10(v.y * SR_INV, (e + 1u) * 16u);
      o.z = draw10(v.z * SR_INV, (e + 2u) * 16u);
      o.w = draw10(v.w * SR_INV, (e + 3u) * 16u);
      gout[i] = o;
    }
    for (int i = n4 * 4 + tid; i < n; i += BLOCK) {
      out[i] = draw10(in[i] * SR_INV, (unsigned)i * 16u);
    }
  }
}

extern "C" void kernel_launch(void* const* d_in, const int* in_sizes, int n_in,
                              void* d_out, int out_size, void* d_ws, size_t ws_size,
                              hipStream_t stream) {
  (void)n_in; (void)out_size; (void)d_ws; (void)ws_size;
  const float* in = (const float*)d_in[0];
  float* out = (float*)d_out;
  const int n = in_sizes[0];      // 8*50*10000*2 = 8,000,000
  const int n4 = n / 4;           // 2,000,000 float4s -> 15625 tiles of 128
  int ntiles = n4 / BLOCK;
  int grid = ntiles < MAXGRID ? (ntiles > 0 ? ntiles : 1) : MAXGRID;
  spike_binomial_kernel<<<grid, BLOCK, 0, stream>>>(in, out, n, n4);
}